// MambaEventDetector_81183471829335
// MI455X (gfx1250) — compile-verified
//
#include <hip/hip_runtime.h>
#include <hip/hip_bf16.h>

// ---------------------------------------------------------------------------
// MI455X (gfx1250, wave32) Mamba forward.
// All GEMMs use native fp32 WMMA: V_WMMA_F32_16X16X4_F32 (D = A*B + C).
// Each wave owns a 64x16 output slab (4 stacked 16x16 WMMA tiles); the k-loop
// is 2-deep software-pipelined so every WMMA group runs with ~6 newer loads
// still in flight (s_wait_loadcnt <= 6, never a full drain).
// ---------------------------------------------------------------------------

typedef __attribute__((ext_vector_type(2))) float v2f;
typedef __attribute__((ext_vector_type(8))) float v8f;

#define D_MODEL 512
#define D_INNER 1024
#define D_STATE 16
#define D_CONV  4
#define DT_RANK 32
#define BATCH   2
#define SEQ     2048
#define NROWS   (BATCH * SEQ)   // 4096
#define EPS     1e-5f
#define MSUB    4               // 16-row subtiles per wave (64 rows total)

__device__ __forceinline__ float silu_f(float x) {
    return x / (1.0f + __expf(-x));
}
__device__ __forceinline__ float softplus_f(float x) {
    // numerically stable softplus, matches jax.nn.softplus
    return (x > 20.0f) ? x : log1pf(__expf(x));
}

// Load one k-step's A (4 subtile fragments) + B fragment into registers.
__device__ __forceinline__ void load_frags(
    const float* __restrict__ Ap, const float* __restrict__ Bp,
    int ldb, size_t Asub, v2f* a, v2f& b)
{
    b.x = Bp[0];
    b.y = Bp[ldb];
    #pragma unroll
    for (int s = 0; s < MSUB; ++s) {
        a[s].x = Ap[s * Asub + 0];
        a[s].y = Ap[s * Asub + 1];
    }
}

// Issue the 4 WMMAs for one k-step.
__device__ __forceinline__ void wmma_group(const v2f* a, v2f b, v8f* acc)
{
    #pragma unroll
    for (int s = 0; s < MSUB; ++s)
        acc[s] = __builtin_amdgcn_wmma_f32_16x16x4_f32(
            false, a[s], false, b, (short)0, acc[s], false, false);
}

// ---------------------------------------------------------------------------
// fp32 WMMA GEMM:  C[M,N] = act( A[M,K] @ B[K,N] + bias ) + residual
// M multiple of 64; N multiple of 16; K multiple of 8 (so K/4 is even).
// One 64x16 slab per wave, 8 waves / block.  act: 0 = identity, 1 = softplus.
// ---------------------------------------------------------------------------
__global__ __launch_bounds__(256) void gemm_wmma_f32(
    const float* __restrict__ A, int lda,
    const float* __restrict__ B, int ldb,
    float* C, int ldc,
    int M, int N, int K,
    const float* __restrict__ bias,
    const float* residual, int ldr,
    int act)
{
    const int tilesN = N >> 4;
    const int tilesM = M >> 6;                    // 64-row slabs
    const int wave = (int)((blockIdx.x * blockDim.x + threadIdx.x) >> 5);
    if (wave >= tilesM * tilesN) return;          // whole-wave exit: EXEC all-1s

    const int tm = wave / tilesN;
    const int tn = wave % tilesN;
    const int lane  = threadIdx.x & 31;
    const int lhalf = lane >> 4;                  // 0: lanes 0-15, 1: lanes 16-31
    const int l16   = lane & 15;
    const int col   = tn * 16 + l16;

    // A fragment (16x4 fp32): lane row = <tile base>+l16 for both halves,
    // lanes 0-15 carry K = {k0,k0+1}, lanes 16-31 carry K = {k0+2,k0+3}.
    const float* Ap = A + (size_t)(tm * 64 + l16) * lda + (lhalf << 1);
    // B fragment (4x16 fp32): v0 = row k0 (+2 upper half), v1 = row k0+1.
    const float* Bp = B + (size_t)(lhalf << 1) * ldb + col;
    const size_t Asub  = (size_t)16 * lda;        // subtile row stride
    const size_t Bstep = (size_t)4 * ldb;

    v8f acc[MSUB];
    #pragma unroll
    for (int s = 0; s < MSUB; ++s)
        acc[s] = (v8f){0.f, 0.f, 0.f, 0.f, 0.f, 0.f, 0.f, 0.f};

    // ---- 2-deep software pipeline: preload k-steps 0 and 1 -----------------
    v2f a_buf0[MSUB], a_buf1[MSUB];
    v2f b_buf0, b_buf1;
    load_frags(Ap, Bp, ldb, Asub, a_buf0, b_buf0);
    Ap += 4; Bp += Bstep;
    load_frags(Ap, Bp, ldb, Asub, a_buf1, b_buf1);

    // Steady state: consume stage s, then issue loads for stage s+2 into the
    // same buffer.  nk = K/4 is even for all K used here, so the hand-unrolled
    // pair keeps buffer parity compile-time constant (no indexed reg arrays).
    const int nk = K >> 2;
    for (int s = 0; s < nk - 2; s += 2) {
        wmma_group(a_buf0, b_buf0, acc);
        Ap += 4; Bp += Bstep;
        load_frags(Ap, Bp, ldb, Asub, a_buf0, b_buf0);
        __builtin_prefetch(Ap + 16, 0, 3);            // global_prefetch_b8
        __builtin_prefetch(Bp + 2 * Bstep, 0, 3);

        wmma_group(a_buf1, b_buf1, acc);
        Ap += 4; Bp += Bstep;
        load_frags(Ap, Bp, ldb, Asub, a_buf1, b_buf1);
    }
    // Drain: last two k-steps.
    wmma_group(a_buf0, b_buf0, acc);
    wmma_group(a_buf1, b_buf1, acc);

    const float bv = bias ? bias[col] : 0.0f;
    // D layout: VGPR j holds row <subtile base> + j + 8*lhalf, col as above.
    #pragma unroll
    for (int s = 0; s < MSUB; ++s) {
        #pragma unroll
        for (int j = 0; j < 8; ++j) {
            const int row = tm * 64 + s * 16 + j + (lhalf << 3);
            float v = acc[s][j] + bv;
            if (act == 1) v = softplus_f(v);
            if (residual) v += residual[(size_t)row * ldr + col];
            C[(size_t)row * ldc + col] = v;
        }
    }
}

// ---------------------------------------------------------------------------
// RMSNorm: one 256-thread block per row of `ncols` (== 512).
// ---------------------------------------------------------------------------
__global__ __launch_bounds__(256) void rmsnorm_kernel(
    const float* __restrict__ x, const float* __restrict__ w,
    float* __restrict__ out, int ncols)
{
    __shared__ float sred[8];
    __shared__ float sscale;
    const int row = blockIdx.x;
    const float* xr = x + (size_t)row * ncols;

    float ss = 0.0f;
    for (int c = threadIdx.x; c < ncols; c += 256) { float v = xr[c]; ss += v * v; }
    #pragma unroll
    for (int o = 16; o > 0; o >>= 1) ss += __shfl_down(ss, o, 32);
    if ((threadIdx.x & 31) == 0) sred[threadIdx.x >> 5] = ss;
    __syncthreads();
    if (threadIdx.x == 0) {
        float t = 0.0f;
        #pragma unroll
        for (int i = 0; i < 8; ++i) t += sred[i];
        sscale = rsqrtf(t / (float)ncols + EPS);
    }
    __syncthreads();
    const float sc = sscale;
    for (int c = threadIdx.x; c < ncols; c += 256)
        out[(size_t)row * ncols + c] = xr[c] * sc * w[c];
}

// ---------------------------------------------------------------------------
// Fused causal depthwise conv (k=4) + bias + SiLU on the xi half of xz,
// writing xc; also SiLU(z) applied in place on the z half of xz.
// One thread per (b,t,d) element, d in [0, D_INNER).
// ---------------------------------------------------------------------------
__global__ __launch_bounds__(256) void conv_silu_kernel(
    float* xz,                         // [NROWS, 2*D_INNER] (z half mutated)
    const float* __restrict__ cw,      // [D_INNER, D_CONV]
    const float* __restrict__ cb,      // [D_INNER]
    float* __restrict__ xc)            // [NROWS, D_INNER]
{
    const size_t idx = (size_t)blockIdx.x * blockDim.x + threadIdx.x;
    if (idx >= (size_t)NROWS * D_INNER) return;
    const int d  = (int)(idx & (D_INNER - 1));
    const size_t bt = idx >> 10;               // row in [0, NROWS)
    const int t = (int)(bt & (SEQ - 1));
    const size_t bbase = bt - t;               // b * SEQ

    float s = cb[d];
    #pragma unroll
    for (int j = 0; j < D_CONV; ++j) {
        const int tt = t - (D_CONV - 1) + j;
        if (tt >= 0)
            s += cw[d * D_CONV + j] * xz[(bbase + tt) * (size_t)(2 * D_INNER) + d];
    }
    xc[bt * D_INNER + d] = silu_f(s);

    const size_t zi = bt * (size_t)(2 * D_INNER) + D_INNER + d;
    xz[zi] = silu_f(xz[zi]);
}

// ---------------------------------------------------------------------------
// Selective scan. One block per batch, one thread per channel d (1024 threads).
// 16 states per thread in registers; B_t / C_t (16+16 floats) staged via
// double-buffered LDS (one barrier per timestep).
// Fuses the D-skip and the * SiLU(z) gate: ym = (y + xc*D) * silu_z.
// ---------------------------------------------------------------------------
__global__ __launch_bounds__(1024) void scan_kernel(
    const float* __restrict__ xc,      // [NROWS, D_INNER]
    const float* __restrict__ delta,   // [NROWS, D_INNER]
    const float* __restrict__ dbc,     // [NROWS, 80]; B at col 32, C at col 48
    const float* __restrict__ A_log,   // [D_INNER, D_STATE] (this layer)
    const float* __restrict__ Dskip,   // [D_INNER]
    const float* __restrict__ sz,      // xz buffer: silu(z) at col D_INNER
    float* __restrict__ ym)            // [NROWS, D_INNER]
{
    __shared__ float BC[2][2 * D_STATE];
    const int b = blockIdx.x;
    const int d = threadIdx.x;

    float Arow[D_STATE];
    #pragma unroll
    for (int s = 0; s < D_STATE; ++s) Arow[s] = -__expf(A_log[d * D_STATE + s]);
    const float Dv = Dskip[d];

    float h[D_STATE];
    #pragma unroll
    for (int s = 0; s < D_STATE; ++s) h[s] = 0.0f;

    for (int t = 0; t < SEQ; ++t) {
        const size_t r = (size_t)b * SEQ + t;
        const int slot = t & 1;
        if (threadIdx.x < 2 * D_STATE)
            BC[slot][threadIdx.x] = dbc[r * 80 + DT_RANK + threadIdx.x];
        __syncthreads();

        const float dlt = delta[r * D_INNER + d];
        const float xv  = xc[r * D_INNER + d];
        const float dx  = dlt * xv;
        float y = 0.0f;
        #pragma unroll
        for (int s = 0; s < D_STATE; ++s) {
            h[s] = __expf(dlt * Arow[s]) * h[s] + dx * BC[slot][s];
            y += h[s] * BC[slot][D_STATE + s];
        }
        ym[r * D_INNER + d] =
            (y + xv * Dv) * sz[r * (size_t)(2 * D_INNER) + D_INNER + d];
    }
}

// ---------------------------------------------------------------------------
// Final rmsnorm + head: tokens = rmsnorm(x, norm_f); logits = tokens.head_w+b.
// One block per row.
// ---------------------------------------------------------------------------
__global__ __launch_bounds__(256) void head_kernel(
    const float* __restrict__ x, const float* __restrict__ norm_f,
    const float* __restrict__ head_w, const float* __restrict__ head_b,
    float* __restrict__ logits, float* __restrict__ tokens)
{
    __shared__ float sred[8];
    __shared__ float sscale;
    const int row = blockIdx.x;
    const float* xr = x + (size_t)row * D_MODEL;

    float ss = 0.0f;
    for (int c = threadIdx.x; c < D_MODEL; c += 256) { float v = xr[c]; ss += v * v; }
    #pragma unroll
    for (int o = 16; o > 0; o >>= 1) ss += __shfl_down(ss, o, 32);
    if ((threadIdx.x & 31) == 0) sred[threadIdx.x >> 5] = ss;
    __syncthreads();
    if (threadIdx.x == 0) {
        float t = 0.0f;
        #pragma unroll
        for (int i = 0; i < 8; ++i) t += sred[i];
        sscale = rsqrtf(t / (float)D_MODEL + EPS);
    }
    __syncthreads();
    const float sc = sscale;

    float dot = 0.0f;
    for (int c = threadIdx.x; c < D_MODEL; c += 256) {
        const float tok = xr[c] * sc * norm_f[c];
        tokens[(size_t)row * D_MODEL + c] = tok;
        dot += tok * head_w[c];
    }
    #pragma unroll
    for (int o = 16; o > 0; o >>= 1) dot += __shfl_down(dot, o, 32);
    __syncthreads();                       // sred safe to reuse now
    if ((threadIdx.x & 31) == 0) sred[threadIdx.x >> 5] = dot;
    __syncthreads();
    if (threadIdx.x == 0) {
        float t = 0.0f;
        #pragma unroll
        for (int i = 0; i < 8; ++i) t += sred[i];
        logits[row] = t + head_b[0];
    }
}

// ---------------------------------------------------------------------------
extern "C" void kernel_launch(void* const* d_in, const int* in_sizes, int n_in,
                              void* d_out, int out_size, void* d_ws, size_t ws_size,
                              hipStream_t stream) {
    (void)in_sizes; (void)n_in; (void)out_size; (void)ws_size;

    const float* features = (const float*)d_in[0];   // [2,2048,512]
    const float* W_in     = (const float*)d_in[1];   // [4,512,2048]
    const float* conv_w   = (const float*)d_in[2];   // [4,1024,4]
    const float* conv_b   = (const float*)d_in[3];   // [4,1024]
    const float* W_x      = (const float*)d_in[4];   // [4,1024,80]
    const float* W_dt     = (const float*)d_in[5];   // [4,32,1024]
    const float* b_dt     = (const float*)d_in[6];   // [4,1024]
    const float* A_log    = (const float*)d_in[7];   // [4,1024,16]
    const float* Dskip    = (const float*)d_in[8];   // [4,1024]
    const float* W_out    = (const float*)d_in[9];   // [4,1024,512]
    const float* norm_w   = (const float*)d_in[10];  // [4,512]
    const float* norm_f   = (const float*)d_in[11];  // [512]
    const float* head_w   = (const float*)d_in[12];  // [512]
    const float* head_b   = (const float*)d_in[13];  // [1]

    // Workspace carve-up (floats): ~97 MB total, L2-resident on MI455X.
    float* ws = (float*)d_ws;
    float* buf_x     = ws;                                   // 4096*512
    float* buf_xn    = buf_x     + (size_t)NROWS * D_MODEL;  // 4096*512
    float* buf_xz    = buf_xn    + (size_t)NROWS * D_MODEL;  // 4096*2048
    float* buf_xc    = buf_xz    + (size_t)NROWS * 2 * D_INNER; // 4096*1024
    float* buf_dbc   = buf_xc    + (size_t)NROWS * D_INNER;  // 4096*80
    float* buf_delta = buf_dbc   + (size_t)NROWS * 80;       // 4096*1024
    float* buf_ym    = buf_delta + (size_t)NROWS * D_INNER;  // 4096*1024

    auto gemm = [&](const float* A, int lda, const float* B, int ldb,
                    float* C, int ldc, int M, int N, int K,
                    const float* bias, const float* res, int ldr, int act) {
        const int tiles  = (M >> 6) * (N >> 4);            // 64x16 slabs
        const int blocks = (tiles + 7) >> 3;               // 8 waves / block
        gemm_wmma_f32<<<blocks, 256, 0, stream>>>(
            A, lda, B, ldb, C, ldc, M, N, K, bias, res, ldr, act);
    };

    const float* x_ptr = features;
    for (int i = 0; i < 4; ++i) {
        // 1) rmsnorm
        rmsnorm_kernel<<<NROWS, 256, 0, stream>>>(
            x_ptr, norm_w + (size_t)i * D_MODEL, buf_xn, D_MODEL);
        // 2) in-projection: xz = xn @ W_in[i]   (4096 x 512 x 2048)
        gemm(buf_xn, D_MODEL, W_in + (size_t)i * D_MODEL * 2 * D_INNER, 2 * D_INNER,
             buf_xz, 2 * D_INNER, NROWS, 2 * D_INNER, D_MODEL,
             nullptr, nullptr, 0, 0);
        // 3) causal dwconv + SiLU, and SiLU(z) in place
        conv_silu_kernel<<<(NROWS * D_INNER + 255) / 256, 256, 0, stream>>>(
            buf_xz, conv_w + (size_t)i * D_INNER * D_CONV,
            conv_b + (size_t)i * D_INNER, buf_xc);
        // 4) x-projection: dbc = xc @ W_x[i]   (4096 x 1024 x 80)
        gemm(buf_xc, D_INNER, W_x + (size_t)i * D_INNER * 80, 80,
             buf_dbc, 80, NROWS, 80, D_INNER, nullptr, nullptr, 0, 0);
        // 5) dt-projection + softplus: delta = softplus(dt @ W_dt[i] + b_dt[i])
        gemm(buf_dbc, 80, W_dt + (size_t)i * DT_RANK * D_INNER, D_INNER,
             buf_delta, D_INNER, NROWS, D_INNER, DT_RANK,
             b_dt + (size_t)i * D_INNER, nullptr, 0, /*softplus*/1);
        // 6) selective scan (fused D-skip and * SiLU(z))
        scan_kernel<<<BATCH, D_INNER, 0, stream>>>(
            buf_xc, buf_delta, buf_dbc,
            A_log + (size_t)i * D_INNER * D_STATE,
            Dskip + (size_t)i * D_INNER, buf_xz, buf_ym);
        // 7) out-projection + residual: x = ym @ W_out[i] + x
        gemm(buf_ym, D_INNER, W_out + (size_t)i * D_INNER * D_MODEL, D_MODEL,
             buf_x, D_MODEL, NROWS, D_MODEL, D_INNER,
             nullptr, x_ptr, D_MODEL, 0);
        x_ptr = buf_x;
    }

    // Final rmsnorm + head.  d_out = [logits (4096) | tokens (4096*512)]
    float* logits = (float*)d_out;
    float* tokens = (float*)d_out + NROWS;
    head_kernel<<<NROWS, 256, 0, stream>>>(
        buf_x, norm_f, head_w, head_b, logits, tokens);
}